// NASP_11579231830854
// MI455X (gfx1250) — compile-verified
//
#include <hip/hip_runtime.h>
#include <hip/hip_bf16.h>
#include <stdint.h>

typedef __attribute__((ext_vector_type(2))) float v2f;
typedef __attribute__((ext_vector_type(4))) float v4f;
typedef __attribute__((ext_vector_type(8))) float v8f;

#define N_COLS  22
#define N_PAIRS 231
#define EMB_NUM 12
#define DD      64
#define ROWS    16          // batch rows per block (WMMA M)
#define WAVES   4           // 128 threads, wave32
#define E_STRIDE (N_COLS * DD)   // 1408 floats per row

// Dynamic LDS layout (floats):
//   E    : [ROWS][N_COLS*DD]            = 22528
//   Weff : [WAVES][4 chunks][2 out][64] =  2048
//   acc  : [16][16]                     =   256
#define LDS_E_FLOATS   (ROWS * E_STRIDE)
#define LDS_W_FLOATS   (WAVES * 4 * 2 * DD)
#define LDS_ACC_FLOATS (16 * 16)
#define LDS_TOTAL_FLOATS (LDS_E_FLOATS + LDS_W_FLOATS + LDS_ACC_FLOATS)

__global__ __launch_bounds__(ROWS * WAVES * 2, 1)
void nasp_wmma_kernel(const int* __restrict__ feats,
                      const float* __restrict__ tables,
                      const float* __restrict__ Wsmall,   // [231][4][2][64]
                      const float* __restrict__ Wcat,     // [231][2][128]
                      const float* __restrict__ arch,     // [231][5]
                      float* __restrict__ out)            // [B][2]
{
    extern __shared__ float lds[];
    float* E    = lds;                          // [ROWS][1408]
    float* Weff = lds + LDS_E_FLOATS;           // [WAVES][4][2][64]
    float* acc  = Weff + LDS_W_FLOATS;          // [16][16]

    const int tid  = threadIdx.x;
    const int lane = tid & 31;
    const int wave = tid >> 5;
    const int b0   = blockIdx.x * ROWS;

    // ---- zero output accumulation tile ----
    for (int t = tid; t < LDS_ACC_FLOATS; t += blockDim.x) acc[t] = 0.0f;

    // ---- gather embedding tile via async global->LDS DMA (16B per lane-op) ----
    // E[row][c][d0..d0+3] = tables[c][feats[b0+row][c]][d0..d0+3]
    {
        const int NGROUPS = ROWS * N_COLS * 16;          // 5632, divisible by 128
        for (int g = tid; g < NGROUPS; g += ROWS * WAVES * 2) {
            int row = g / (N_COLS * 16);
            int rc  = g - row * (N_COLS * 16);
            int c   = rc >> 4;
            int dg  = rc & 15;                            // 4-float group within D
            int idx = feats[(b0 + row) * N_COLS + c];
            uint64_t src = (uint64_t)(uintptr_t)(tables + (c * EMB_NUM + idx) * DD + dg * 4);
            unsigned lds_off = (unsigned)((row * E_STRIDE + c * DD + dg * 4) * sizeof(float));
            asm volatile("global_load_async_to_lds_b128 %0, %1, off"
                         :: "v"(lds_off), "v"(src) : "memory");
        }
        asm volatile("s_wait_asynccnt 0x0" ::: "memory");
    }
    __syncthreads();

    // ---- per-wave WMMA accumulator over the pair loop ----
    v8f C = {};

    float* myW = Weff + wave * (4 * 2 * DD);
    const int row = lane & 15;     // A fragment: M row
    const int kl  = lane >> 4;     // A fragment: K sub-group (K = 4s + 2*kl + {0,1})
    const int n   = lane & 15;     // B fragment: N column (only n<2 carries weight)

    // staging role of this lane: contiguous 16-float run
    const int st_chunk = lane >> 3;          // 0..3
    const int st_o     = (lane >> 2) & 1;    // 0..1
    const int st_d0    = (lane & 3) * 16;    // 0,16,32,48

    for (int p = wave; p < N_PAIRS; p += WAVES) {
        // force wave-uniformity so (i,j) + weight addressing goes scalar/SMEM
        const int pu = __builtin_amdgcn_readfirstlane(p);

        // triu indices (i,j) for pair pu (scalar loop)
        int i = 0, rem = pu;
        while (rem >= (N_COLS - 1 - i)) { rem -= (N_COLS - 1 - i); ++i; }
        const int j = i + 1 + rem;

        const float a0 = arch[pu * 5 + 0];
        const float a1 = arch[pu * 5 + 1];
        const float a2 = arch[pu * 5 + 2];
        const float a3 = arch[pu * 5 + 3];
        const float a4 = arch[pu * 5 + 4];

        // prefetch next pair's weights while we work on this one
        if (pu + WAVES < N_PAIRS) {
            __builtin_prefetch(Wsmall + (pu + WAVES) * 4 * 2 * DD, 0, 0);
            __builtin_prefetch(Wcat + (pu + WAVES) * 2 * 128, 0, 0);
        }

        // ---- stage effective weights for this pair: Weff[chunk][o][d], 512 floats ----
        // chunk 0: P     -> a0*Ws0 + a3*Ws3 + a4*Wc[:, :64]
        // chunk 1: Q     -> a0*Ws0 + a3*Ws3 + a4*Wc[:, 64:]
        // chunk 2: P*Q   -> a1*Ws1
        // chunk 3: max   -> a2*Ws2 - a3*Ws3        (min = P+Q-max folded into 0,1,3)
        {
            const float* Ws  = Wsmall + pu * 4 * 2 * DD;
            const float* Wcp = Wcat + pu * 2 * 128 + st_o * 128 + (st_chunk == 1 ? 64 : 0);
            const bool lin = (st_chunk < 2);
            const float c0 = lin ? a0 : 0.0f;
            const float c1 = (st_chunk == 2) ? a1 : 0.0f;
            const float c2 = (st_chunk == 3) ? a2 : 0.0f;
            const float c3 = lin ? a3 : ((st_chunk == 3) ? -a3 : 0.0f);
            const float c4 = lin ? a4 : 0.0f;
            #pragma unroll
            for (int t = 0; t < 4; ++t) {
                const int d = st_d0 + 4 * t;
                v4f w0 = *(const v4f*)(Ws + (0 * 2 + st_o) * DD + d);
                v4f w1 = *(const v4f*)(Ws + (1 * 2 + st_o) * DD + d);
                v4f w2 = *(const v4f*)(Ws + (2 * 2 + st_o) * DD + d);
                v4f w3 = *(const v4f*)(Ws + (3 * 2 + st_o) * DD + d);
                v4f wc = *(const v4f*)(Wcp + d);
                v4f val = c0 * w0 + c1 * w1 + c2 * w2 + c3 * w3 + c4 * wc;
                *(v4f*)(myW + st_chunk * 128 + st_o * DD + d) = val;
            }
        }
        // Same-wave LDS ops are in-order; barrier pins the schedule.
        __builtin_amdgcn_wave_barrier();

        const float* Prow = E + row * E_STRIDE + i * DD;
        const float* Qrow = E + row * E_STRIDE + j * DD;

        #pragma unroll
        for (int chunk = 0; chunk < 4; ++chunk) {
            const float* Wb = myW + chunk * 128 + n * DD;  // [o=n][d], valid for n<2
            #pragma unroll
            for (int s = 0; s < 16; ++s) {
                const int d0 = 4 * s + 2 * kl;
                v2f P2 = *(const v2f*)(Prow + d0);
                v2f Q2 = *(const v2f*)(Qrow + d0);
                v2f a;
                if (chunk == 0)      a = P2;
                else if (chunk == 1) a = Q2;
                else if (chunk == 2) a = P2 * Q2;
                else { a.x = fmaxf(P2.x, Q2.x); a.y = fmaxf(P2.y, Q2.y); }
                v2f b; b.x = 0.0f; b.y = 0.0f;
                if (n < 2) b = *(const v2f*)(Wb + d0);
                // D(16x16 f32) = A(16x4 f32) * B(4x16 f32) + C ; chained accumulation
                C = __builtin_amdgcn_wmma_f32_16x16x4_f32(
                        false, a, false, b, (short)0, C, false, false);
            }
        }
    }

    // ---- cross-wave reduction of the 16x16 accumulator via LDS float atomics ----
    // C layout: VGPR r, lane l -> M = r + 8*(l>>4), N = l&15
    {
        const int m_hi = (lane >> 4) * 8;
        const int nn   = lane & 15;
        #pragma unroll
        for (int r = 0; r < 8; ++r) {
            atomicAdd(&acc[(m_hi + r) * 16 + nn], C[r]);
        }
    }
    __syncthreads();

    // ---- write 16 rows x 2 outputs ----
    if (tid < 32) {
        int m = tid >> 1;
        int o = tid & 1;
        out[(b0 + m) * 2 + o] = acc[m * 16 + o];
    }
}

extern "C" void kernel_launch(void* const* d_in, const int* in_sizes, int n_in,
                              void* d_out, int out_size, void* d_ws, size_t ws_size,
                              hipStream_t stream) {
    const int*   feats  = (const int*)d_in[0];
    const float* tables = (const float*)d_in[1];
    const float* Wsmall = (const float*)d_in[2];
    const float* Wcat   = (const float*)d_in[3];
    const float* arch   = (const float*)d_in[4];
    float*       out    = (float*)d_out;

    const int B = in_sizes[0] / N_COLS;       // 4096
    const int grid = B / ROWS;                // 256 blocks of 16 rows
    const size_t shmem = LDS_TOTAL_FLOATS * sizeof(float);  // ~97 KB dynamic LDS

    nasp_wmma_kernel<<<grid, ROWS * WAVES * 2, shmem, stream>>>(
        feats, tables, Wsmall, Wcat, arch, out);
}